// GraphSageConv_xn_only_76192719831692
// MI455X (gfx1250) — compile-verified
//
#include <hip/hip_runtime.h>

#define H 128
#define LDSS 260   // 256 + 4 dword pad: keeps 16B alignment, kills LDS bank conflicts

typedef float v2f __attribute__((ext_vector_type(2)));
typedef float v4f __attribute__((ext_vector_type(4)));
typedef float v8f __attribute__((ext_vector_type(8)));

// ---------------- zero ----------------
__global__ void zero_kernel(float* __restrict__ p, long n4) {
  long i = (long)blockIdx.x * blockDim.x + threadIdx.x;
  if (i < n4) {
    v4f z = {0.f, 0.f, 0.f, 0.f};
    ((v4f*)p)[i] = z;
  }
}

// ---------------- edge aggregation: agg[dst] += h[src] ----------------
// one wave32 per edge, float4 per lane (32*4 = 128 feats), f32 atomics into L2.
// edge index is wave-uniform: force it into an SGPR so src/dst become scalar
// loads and the gather/scatter bases are uniform (global_load/atomic + saddr).
__global__ __launch_bounds__(256)
void aggregate_kernel(const float* __restrict__ h,
                      const int* __restrict__ src,
                      const int* __restrict__ dst,
                      float* __restrict__ agg, int n_edges) {
  int e = blockIdx.x * 8 + (threadIdx.x >> 5);   // 8 waves/block, 1 edge/wave
  if (e >= n_edges) return;                       // wave-uniform guard
  e = __builtin_amdgcn_readfirstlane(e);
  const int s = __builtin_amdgcn_readfirstlane(src[e]);
  const int d = __builtin_amdgcn_readfirstlane(dst[e]);
  const int l = (int)(threadIdx.x & 31);
  v4f v = *(const v4f*)(h + (size_t)s * H + l * 4);
  float* p = agg + (size_t)d * H + l * 4;
  unsafeAtomicAdd(p + 0, v.x);
  unsafeAtomicAdd(p + 1, v.y);
  unsafeAtomicAdd(p + 2, v.z);
  unsafeAtomicAdd(p + 3, v.w);
}

// ---------------- SAGE GEMM: out = relu([A0|A1] @ W + bias) ----------------
// A0: [n, K0], A1: [n, K1] (K1 may be 0), W: [K0+K1, 128], out: [n, 128]
// block = 256 threads = 8 waves; block computes 16 rows x 128 cols;
// wave w computes the 16x16 tile at columns [16w, 16w+16) via f32 WMMA.
__global__ __launch_bounds__(256)
void sage_gemm_kernel(const float* __restrict__ A0, const float* __restrict__ A1,
                      const float* __restrict__ W, const float* __restrict__ bias,
                      float* __restrict__ out, int n_nodes, int K0, int K1,
                      int do_relu) {
  __shared__ float lds[16 * LDSS];
  const int K = K0 + K1;
  const int m0 = blockIdx.x * 16;
  const int tid = threadIdx.x;
  const int lane = tid & 31;
  const int wave = tid >> 5;

  // warm the weight tile this wave will stream (emits global_prefetch_b8)
  __builtin_prefetch(W + (size_t)wave * 16, 0, 2);

  // cooperative, coalesced load of the 16 x K A-tile into padded LDS
  const int nvec = K >> 2;  // float4s per row
  for (int i = tid; i < 16 * nvec; i += 256) {
    int r = i / nvec;
    int c = (i - r * nvec) * 4;
    int node = m0 + r;
    v4f v = {0.f, 0.f, 0.f, 0.f};
    if (node < n_nodes) {
      if (c < K0) v = *(const v4f*)(A0 + (size_t)node * K0 + c);
      else        v = *(const v4f*)(A1 + (size_t)node * K1 + (c - K0));
    }
    *(v4f*)(&lds[r * LDSS + c]) = v;
  }
  __syncthreads();

  // WMMA fragment addressing (ISA 7.12.2, 16x4 f32 A / 4x16 B layouts):
  //   A: lane%16 = row M, lane/16 selects K pair {0,1} vs {2,3}
  //   B: lane%16 = col N, lane/16 selects K rows {0,1} vs {2,3}
  const int n = wave * 16 + (lane & 15);
  const int khalf = (lane >> 4) << 1;  // 0 or 2
  const float* arow = &lds[(lane & 15) * LDSS + khalf];

  v8f c;
  {
    float bv = bias[n];  // bias depends only on N -> same in all 8 acc VGPRs
#pragma unroll
    for (int i = 0; i < 8; ++i) c[i] = bv;
  }

#pragma unroll 8
  for (int k = 0; k < K; k += 4) {
    v2f a = *(const v2f*)(arow + k);          // conflict-free ds_load_2addr_b64
    v2f b;
    b.x = W[(k + khalf) * H + n];             // L2-resident weights
    b.y = W[(k + khalf + 1) * H + n];
    c = __builtin_amdgcn_wmma_f32_16x16x4_f32(
        /*neg_a=*/false, a, /*neg_b=*/false, b,
        /*c_mod=*/(short)0, c, /*reuse_a=*/false, /*reuse_b=*/false);
  }

  // C/D layout: VGPR r holds M = r (lanes 0-15) / r+8 (lanes 16-31), N = lane%16
  const int mbase = m0 + ((lane >> 4) << 3);
#pragma unroll
  for (int r = 0; r < 8; ++r) {
    int row = mbase + r;
    if (row < n_nodes) {
      float v = c[r];
      if (do_relu) v = fmaxf(v, 0.0f);
      out[(size_t)row * H + n] = v;
    }
  }
}

// ---------------- final [128] -> 1 matvec: out[i] = h[i,:] . w + b ----------------
__global__ __launch_bounds__(256)
void matvec_kernel(const float* __restrict__ h, const float* __restrict__ w,
                   const float* __restrict__ b, float* __restrict__ out,
                   int n_nodes) {
  int node = blockIdx.x * 8 + (threadIdx.x >> 5);
  int lane = threadIdx.x & 31;
  if (node >= n_nodes) return;                 // wave-uniform guard
  node = __builtin_amdgcn_readfirstlane(node); // scalar row base
  v4f hv = *(const v4f*)(h + (size_t)node * H + lane * 4);
  v4f wv = *(const v4f*)(w + lane * 4);
  float s = hv.x * wv.x + hv.y * wv.y + hv.z * wv.z + hv.w * wv.w;
#pragma unroll
  for (int off = 16; off > 0; off >>= 1) s += __shfl_xor(s, off, 32);
  if (lane == 0) out[node] = s + b[0];
}

extern "C" void kernel_launch(void* const* d_in, const int* in_sizes, int n_in,
                              void* d_out, int out_size, void* d_ws, size_t ws_size,
                              hipStream_t stream) {
  const float* node_feat = (const float*)d_in[0];
  // d_in[1] = edge_feat: unused by the reference network
  const int* src = (const int*)d_in[2];
  const int* dst = (const int*)d_in[3];
  const float* W1 = (const float*)d_in[4];
  const float* b1 = (const float*)d_in[5];
  const float* Wm[4] = {(const float*)d_in[6], (const float*)d_in[8],
                        (const float*)d_in[10], (const float*)d_in[12]};
  const float* bm[4] = {(const float*)d_in[7], (const float*)d_in[9],
                        (const float*)d_in[11], (const float*)d_in[13]};
  const float* Wr1 = (const float*)d_in[14];
  const float* br1 = (const float*)d_in[15];
  const float* Wr2 = (const float*)d_in[16];
  const float* br2 = (const float*)d_in[17];
  const float* Wr3 = (const float*)d_in[18];
  const float* br3 = (const float*)d_in[19];

  const int N = in_sizes[0] / H;  // 100000 nodes
  const int E = in_sizes[2];      // 1600000 edges

  float* hA = (float*)d_ws;
  float* hB = hA + (size_t)N * H;
  float* agg = hB + (size_t)N * H;

  const long n4 = (long)N * H / 4;
  dim3 zgrid((unsigned)((n4 + 255) / 256));
  dim3 agrid((unsigned)((E + 7) / 8));
  dim3 ggrid((unsigned)((N + 15) / 16));
  dim3 mgrid((unsigned)((N + 7) / 8));

  auto layer = [&](const float* hin, const float* Wx, const float* bx,
                   float* hout) {
    zero_kernel<<<zgrid, 256, 0, stream>>>(agg, n4);
    aggregate_kernel<<<agrid, 256, 0, stream>>>(hin, src, dst, agg, E);
    sage_gemm_kernel<<<ggrid, 256, 0, stream>>>(hin, agg, Wx, bx, hout, N, H, H,
                                                1);
  };

  layer(node_feat, W1, b1, hA);     // conv1
  layer(hA, Wm[0], bm[0], hB);      // convmid1
  layer(hB, Wm[1], bm[1], hA);      // convmid2
  layer(hA, Wm[2], bm[2], hB);      // convmid3
  layer(hB, Wm[3], bm[3], hA);      // convmid4
  layer(hA, Wr1, br1, hB);          // regression: aggr + concat + Wr1 + relu
  sage_gemm_kernel<<<ggrid, 256, 0, stream>>>(hB, nullptr, Wr2, br2, agg, N, H,
                                              0, 1);  // Wr2 + relu
  matvec_kernel<<<mgrid, 256, 0, stream>>>(agg, Wr3, br3, (float*)d_out, N);
}